// GCNModel_11407433138237
// MI455X (gfx1250) — compile-verified
//
#include <hip/hip_runtime.h>
#include <hip/hip_bf16.h>

typedef float v2f __attribute__((ext_vector_type(2)));
typedef float v8f __attribute__((ext_vector_type(8)));

#define FH 256          // hidden width (fixed 256 for the two big GEMMs)
#define BN_EPS 1e-5f

// ---------------------------------------------------------------------------
// Degree / norm precompute
// ---------------------------------------------------------------------------
__global__ void fill_deg(float* __restrict__ deg, int n) {
    int i = blockIdx.x * blockDim.x + threadIdx.x;
    if (i < n) deg[i] = 1.0f;                        // self-loop weight 1
}

__global__ void deg_accum(float* __restrict__ deg, const long long* __restrict__ col,
                          const float* __restrict__ w, int e) {
    int i = blockIdx.x * blockDim.x + threadIdx.x;
    if (i < e) atomicAdd(&deg[col[i]], w[i]);
}

__global__ void dinv_selfnorm(float* __restrict__ deg_to_dinv, float* __restrict__ sn, int n) {
    int i = blockIdx.x * blockDim.x + threadIdx.x;
    if (i >= n) return;
    float d = deg_to_dinv[i];
    float di = d > 0.0f ? rsqrtf(d) : 0.0f;
    deg_to_dinv[i] = di;                             // in place: deg -> dinv
    sn[i] = di * di;                                 // self-loop norm
}

__global__ void norm_kernel(float* __restrict__ nrm, const float* __restrict__ dinv,
                            const long long* __restrict__ row, const long long* __restrict__ col,
                            const float* __restrict__ w, int e) {
    int i = blockIdx.x * blockDim.x + threadIdx.x;
    if (i < e) nrm[i] = dinv[row[i]] * w[i] * dinv[col[i]];
}

// ---------------------------------------------------------------------------
// fp32 WMMA GEMM: C[M x 256] = A[M x K] @ B[K x 256]
// Block = 256 threads (8 wave32). Block covers 16 rows x 256 cols.
// Wave w owns columns [32w, 32w+32): two 16x16 tiles, two v8f accumulators.
// A 16x32 chunk staged in LDS (row stride 34: even -> 8B-aligned v2f, no
// bank conflicts). Fragment layouts per CDNA5 ISA 7.12.2:
//   A (16x4): a[j] = A[m][2*half + j],   m = lane%16, half = lane/16
//   B (4x16): b[j] = B[2*half + j][n],   n = lane%16
//   C/D     : c[v] = C[v + 8*half][n]
// ---------------------------------------------------------------------------
__global__ __launch_bounds__(256) void gemm_wmma_f32(
    const float* __restrict__ A, const float* __restrict__ B,
    float* __restrict__ C, int M, int K)
{
    __shared__ __align__(16) float As[16 * 34];
    const int tid  = threadIdx.x;
    const int lane = tid & 31;
    const int wave = tid >> 5;       // 0..7
    const int half = lane >> 4;      // 0/1
    const int lm   = lane & 15;      // 0..15
    const int m0   = blockIdx.x * 16;
    const int n0   = wave * 32;

    v8f acc0 = {};
    v8f acc1 = {};

    for (int k0 = 0; k0 < K; k0 += 32) {
        __syncthreads();
        {   // cooperative A stage: 16 rows x 32 cols, 2 floats per thread
            int r = tid >> 4;
            int c = (tid & 15) * 2;
            int rr = m0 + r; if (rr >= M) rr = M - 1;
            v2f av = *(const v2f*)(A + (size_t)rr * K + k0 + c);
            As[r * 34 + c]     = av.x;
            As[r * 34 + c + 1] = av.y;
        }
        __syncthreads();
        #pragma unroll
        for (int ks = 0; ks < 32; ks += 4) {
            v2f a = *(const v2f*)(&As[lm * 34 + ks + 2 * half]);
            int kr = k0 + ks + 2 * half;
            const float* bp = B + (size_t)kr * FH + n0 + lm;
            v2f b0; b0.x = bp[0];  b0.y = bp[FH];
            v2f b1; b1.x = bp[16]; b1.y = bp[FH + 16];
            acc0 = __builtin_amdgcn_wmma_f32_16x16x4_f32(
                       false, a, false, b0, (short)0, acc0, false, false);
            acc1 = __builtin_amdgcn_wmma_f32_16x16x4_f32(
                       false, a, false, b1, (short)0, acc1, false, false);
        }
    }

    if (m0 + 16 <= M) {
        // fast path: whole tile in range, unguarded coalesced stores
        float* cp = C + (size_t)(m0 + 8 * half) * FH + n0 + lm;
        #pragma unroll
        for (int v = 0; v < 8; ++v) {
            cp[(size_t)v * FH]      = acc0[v];
            cp[(size_t)v * FH + 16] = acc1[v];
        }
    } else {
        #pragma unroll
        for (int v = 0; v < 8; ++v) {
            int m = m0 + v + 8 * half;
            if (m < M) {
                C[(size_t)m * FH + n0 + lm]      = acc0[v];
                C[(size_t)m * FH + n0 + 16 + lm] = acc1[v];
            }
        }
    }
}

// ---------------------------------------------------------------------------
// Aggregation (F = 256)
// ---------------------------------------------------------------------------
__global__ void init_agg256(const float* __restrict__ h, float* __restrict__ out,
                            const float* __restrict__ sn, int total) {
    int i = blockIdx.x * blockDim.x + threadIdx.x;
    if (i < total) out[i] = sn[i >> 8] * h[i];
}

__global__ __launch_bounds__(256) void edge_scatter256(
    const float* __restrict__ h, float* __restrict__ out,
    const long long* __restrict__ row, const long long* __restrict__ col,
    const float* __restrict__ nrm, int e)
{
    // one wave per edge; edge id is wave-uniform -> force scalar path
    int edge = (int)((blockIdx.x * 256 + threadIdx.x) >> 5);
    edge = __builtin_amdgcn_readfirstlane(edge);
    int lane = threadIdx.x & 31;
    if (edge >= e) return;
    long long r = row[edge];
    long long c = col[edge];
    float w = nrm[edge];
    const float* hp = h + (size_t)r * FH;
    float* op = out + (size_t)c * FH;
    #pragma unroll
    for (int j = 0; j < 2; ++j) {
        int f = lane * 4 + j * 128;
        float4 hv = *(const float4*)(hp + f);
        atomicAdd(op + f + 0, w * hv.x);
        atomicAdd(op + f + 1, w * hv.y);
        atomicAdd(op + f + 2, w * hv.z);
        atomicAdd(op + f + 3, w * hv.w);
    }
}

__global__ void bn_relu256(float* __restrict__ x, const float* __restrict__ b,
                           const float* __restrict__ g,  const float* __restrict__ be,
                           const float* __restrict__ m,  const float* __restrict__ v,
                           int total) {
    int i = blockIdx.x * blockDim.x + threadIdx.x;
    if (i >= total) return;
    int f = i & (FH - 1);
    float val = x[i] + b[f];
    val = (val - m[f]) * rsqrtf(v[f] + BN_EPS) * g[f] + be[f];
    x[i] = val > 0.0f ? val : 0.0f;
}

// ---------------------------------------------------------------------------
// Final projection 256 -> 2 (wave-per-node reduction) + aggregation on 2 feats
// ---------------------------------------------------------------------------
__global__ __launch_bounds__(256) void gemm_out2(
    const float* __restrict__ A, const float* __restrict__ W,
    float* __restrict__ H3, int M)
{
    __shared__ float Ws[FH * 2];
    int tid = threadIdx.x;
    Ws[tid]       = W[tid];
    Ws[tid + 256] = W[tid + 256];
    __syncthreads();
    int lane = tid & 31, wave = tid >> 5;
    int node = blockIdx.x * 8 + wave;
    if (node >= M) return;
    float s0 = 0.0f, s1 = 0.0f;
    #pragma unroll
    for (int j = 0; j < 8; ++j) {
        int k = lane + 32 * j;
        float av = A[(size_t)node * FH + k];
        s0 += av * Ws[k * 2];
        s1 += av * Ws[k * 2 + 1];
    }
    #pragma unroll
    for (int off = 16; off > 0; off >>= 1) {
        s0 += __shfl_down(s0, off, 32);
        s1 += __shfl_down(s1, off, 32);
    }
    if (lane == 0) {
        H3[(size_t)node * 2]     = s0;
        H3[(size_t)node * 2 + 1] = s1;
    }
}

__global__ void init_out2(const float* __restrict__ h3, float* __restrict__ out,
                          const float* __restrict__ sn, const float* __restrict__ b3,
                          int total) {
    int i = blockIdx.x * blockDim.x + threadIdx.x;
    if (i >= total) return;
    out[i] = sn[i >> 1] * h3[i] + b3[i & 1];
}

__global__ void edge_scatter2(const float* __restrict__ h3, float* __restrict__ out,
                              const long long* __restrict__ row, const long long* __restrict__ col,
                              const float* __restrict__ nrm, int e) {
    int i = blockIdx.x * blockDim.x + threadIdx.x;
    if (i >= e) return;
    long long r = row[i];
    long long c = col[i];
    float w = nrm[i];
    atomicAdd(&out[c * 2],     w * h3[r * 2]);
    atomicAdd(&out[c * 2 + 1], w * h3[r * 2 + 1]);
}

// ---------------------------------------------------------------------------
// Launcher
// ---------------------------------------------------------------------------
static inline int cdiv(int a, int b) { return (a + b - 1) / b; }

extern "C" void kernel_launch(void* const* d_in, const int* in_sizes, int n_in,
                              void* d_out, int out_size, void* d_ws, size_t ws_size,
                              hipStream_t stream)
{
    const float*      x    = (const float*)d_in[0];
    const long long*  eidx = (const long long*)d_in[1];   // int64 [2, E]
    const float*      ew   = (const float*)d_in[2];
    const float*      W1   = (const float*)d_in[3];
    const float*      b1   = (const float*)d_in[4];
    const float*      g1   = (const float*)d_in[5];
    const float*      be1  = (const float*)d_in[6];
    const float*      m1   = (const float*)d_in[7];
    const float*      v1   = (const float*)d_in[8];
    const float*      W2   = (const float*)d_in[9];
    const float*      b2   = (const float*)d_in[10];
    const float*      g2   = (const float*)d_in[11];
    const float*      be2  = (const float*)d_in[12];
    const float*      m2   = (const float*)d_in[13];
    const float*      v2   = (const float*)d_in[14];
    const float*      W3   = (const float*)d_in[15];
    const float*      b3   = (const float*)d_in[16];

    const int E = in_sizes[2];                 // edge_weight count
    const int N = in_sizes[0] / 512;           // x is [N, 512]
    const long long* row  = eidx;
    const long long* colv = eidx + E;
    float* out = (float*)d_out;                // [N, 2]

    // workspace carve-out (floats)
    float* ws   = (float*)d_ws;
    size_t off  = 0;
    float* h    = ws + off; off += (size_t)N * FH;   // GEMM output buffer
    float* agg  = ws + off; off += (size_t)N * FH;   // aggregation buffer
    float* h3v  = ws + off; off += (size_t)N * 2;
    float* dinv = ws + off; off += (size_t)N;        // deg, then dinv in place
    float* sn   = ws + off; off += (size_t)N;        // self-loop norm
    float* nrm  = ws + off; off += (size_t)E;        // per-edge norm

    // --- degree / norm precompute ---
    fill_deg     <<<cdiv(N, 256), 256, 0, stream>>>(dinv, N);
    deg_accum    <<<cdiv(E, 256), 256, 0, stream>>>(dinv, colv, ew, E);
    dinv_selfnorm<<<cdiv(N, 256), 256, 0, stream>>>(dinv, sn, N);
    norm_kernel  <<<cdiv(E, 256), 256, 0, stream>>>(nrm, dinv, row, colv, ew, E);

    const int total = N * FH;

    // --- layer 1: h = x @ W1 ; aggregate ; bias+BN+ReLU ---
    gemm_wmma_f32 <<<cdiv(N, 16), 256, 0, stream>>>(x, W1, h, N, 512);
    init_agg256   <<<cdiv(total, 256), 256, 0, stream>>>(h, agg, sn, total);
    edge_scatter256<<<cdiv(E, 8), 256, 0, stream>>>(h, agg, row, colv, nrm, E);
    bn_relu256    <<<cdiv(total, 256), 256, 0, stream>>>(agg, b1, g1, be1, m1, v1, total);

    // --- layer 2: h = a1 @ W2 ; aggregate ; bias+BN+ReLU ---
    gemm_wmma_f32 <<<cdiv(N, 16), 256, 0, stream>>>(agg, W2, h, N, 256);
    init_agg256   <<<cdiv(total, 256), 256, 0, stream>>>(h, agg, sn, total);
    edge_scatter256<<<cdiv(E, 8), 256, 0, stream>>>(h, agg, row, colv, nrm, E);
    bn_relu256    <<<cdiv(total, 256), 256, 0, stream>>>(agg, b2, g2, be2, m2, v2, total);

    // --- layer 3: h3 = a2 @ W3 ; aggregate (+b3) into d_out ---
    gemm_out2     <<<cdiv(N, 8), 256, 0, stream>>>(agg, W3, h3v, N);
    init_out2     <<<cdiv(N * 2, 256), 256, 0, stream>>>(h3v, out, sn, b3, N * 2);
    edge_scatter2 <<<cdiv(E, 256), 256, 0, stream>>>(h3v, out, row, colv, nrm, E);
}